// RealAgnosticDensityInjuctedNoScaleNoBiasInteractionGateBlock_21766894256494
// MI455X (gfx1250) — compile-verified
//
#include <hip/hip_runtime.h>
#include <math.h>

// ---------------------------------------------------------------------------
// Problem constants (from the reference)
// ---------------------------------------------------------------------------
#define N_NODES   10000
#define N_EDGES   160000
#define MUL       128
#define N_SPECIES 10
#define MSG_W     2048            // MUL * 16  (sum of 128*d over l, d = 1,3,5,7)

#define RS_MUL  0.08838834764831845f   // 1/sqrt(128)
#define RS_8    0.3535533905932738f    // 1/sqrt(8)
#define RS_64   0.125f                 // 1/sqrt(64)
#define RS_SKIP 0.02795084971874737f   // 1/sqrt(1280)

typedef float v2f __attribute__((ext_vector_type(2)));
typedef float v8f __attribute__((ext_vector_type(8)));

// CDNA5 f32 WMMA: D(16x16) = A(16x4) * B(4x16) + C   (wave32, exact fp32)
__device__ __forceinline__ v8f wmma_f32x4(v2f a, v2f b, v8f c) {
  return __builtin_amdgcn_wmma_f32_16x16x4_f32(
      /*neg_a=*/false, a, /*neg_b=*/false, b,
      /*c_mod=*/(short)0, c, /*reuse_a=*/false, /*reuse_b=*/false);
}

__device__ __forceinline__ v8f v8f_zero() {
  v8f c = {0.f, 0.f, 0.f, 0.f, 0.f, 0.f, 0.f, 0.f};
  return c;
}

__device__ __forceinline__ float silu(float x) {
  return x / (1.0f + __expf(-x));
}

// ---------------------------------------------------------------------------
// K0: zero fill (message accumulator + rho before the atomic scatter)
// ---------------------------------------------------------------------------
__global__ void zero_f32(float* __restrict__ p, size_t n) {
  size_t i = (size_t)blockIdx.x * blockDim.x + threadIdx.x;
  size_t stride = (size_t)gridDim.x * blockDim.x;
  for (; i < n; i += stride) p[i] = 0.0f;
}

// ---------------------------------------------------------------------------
// K1/K5: WMMA GEMM over a strided row view, templated on the interleave D so
// all index math folds to constant-multiply/shift (no runtime idiv chains).
//   Logical A[m][u] = src[(m/D)*srcStride + base + u*D + (m%D)]   (K = 128)
//   C[m][v]*scale  -> dst[(m/D)*dstStride + base + v*D + (m%D)]
// D=1,base=0: plain [M,128]@[128,128] GEMM (node up-projection, Wlin l=0).
// D=2l+1, base=128*l*l: per-l Wlin einsum 'nuk,uv->nvk'.
// One wave computes one 16x16 output tile; all tiles are full (M % 16 == 0).
// ---------------------------------------------------------------------------
template <int D>
__global__ __launch_bounds__(32) void gemm_wmma_strided(
    const float* __restrict__ src, const float* __restrict__ W,
    float* __restrict__ dst, int base,
    int srcStride, int dstStride, float scale) {
  const int lane = threadIdx.x & 31;
  const int l16 = lane & 15;
  const int half = lane >> 4;
  const int row0 = blockIdx.x * 16;
  const int col0 = blockIdx.y * 16;

  const int m  = row0 + l16;
  const int nn = m / D;                 // compile-time-constant divisor
  const int kk = m - nn * D;
  const float* __restrict__ arow = src + (size_t)nn * srcStride + base + kk;
  const float* __restrict__ bcol = W + col0 + l16;

  v8f c = v8f_zero();
  for (int k0 = 0; k0 < 128; k0 += 4) {
    const int u0 = k0 + 2 * half;
    v2f a;
    if constexpr (D == 1) {
      // contiguous pair, 8B aligned (u0 even, rows >= 512B aligned) -> b64
      a = *(const v2f*)(arow + u0);
    } else {
      a = v2f{ arow[(size_t)u0 * D], arow[(size_t)(u0 + 1) * D] };
    }
    v2f b = { bcol[(size_t)u0 * 128], bcol[(size_t)(u0 + 1) * 128] };
    c = wmma_f32x4(a, b, c);
  }

#pragma unroll
  for (int i = 0; i < 8; ++i) {
    const int mo = row0 + i + 8 * half;
    const int no = mo / D;
    const int ko = mo - no * D;
    dst[(size_t)no * dstStride + base + (size_t)(col0 + l16) * D + ko] =
        c[i] * scale;
  }
}

// ---------------------------------------------------------------------------
// K2: fused radial MLP (WMMA) + density + tensor-product scatter.
// One wave handles a tile of 16 edges. tp_w never touches global memory:
// each layer-4 16x16 accumulator fragment is scattered directly into the
// L2-resident message[] via global_atomic_add_f32.
// ---------------------------------------------------------------------------
__global__ __launch_bounds__(32) void radial_scatter(
    const float* __restrict__ edge_feats, const float* __restrict__ edge_attrs,
    const int* __restrict__ edge_index, const float* __restrict__ x,
    const float* __restrict__ Wr1, const float* __restrict__ Wr2,
    const float* __restrict__ Wr3, const float* __restrict__ Wr4,
    const float* __restrict__ Wd,
    float* __restrict__ msg, float* __restrict__ rho) {
  __shared__ float sEF[16][8];     // edge_feats tile
  __shared__ float sSH[16][16];    // spherical harmonics tile
  __shared__ float sXS[16][128];   // gathered sender features
  __shared__ float sHA[16][64];    // MLP activations (ping)
  __shared__ float sHB[16][64];    // MLP activations (pong)
  __shared__ int   sR[16];
  __shared__ int   sS[16];

  const int lane = threadIdx.x;
  const int e0 = blockIdx.x * 16;

  if (lane < 16) {
    sS[lane] = edge_index[e0 + lane];
    sR[lane] = edge_index[N_EDGES + e0 + lane];
  }
  for (int i = lane; i < 16 * 8; i += 32)
    sEF[i >> 3][i & 7] = edge_feats[(size_t)e0 * 8 + i];
  for (int i = lane; i < 16 * 16; i += 32)
    sSH[i >> 4][i & 15] = edge_attrs[(size_t)e0 * 16 + i];
  __syncthreads();

  // gather x[sender] (L2-resident, 5 MB table)
  for (int i = lane; i < 16 * 128; i += 32) {
    const int m = i >> 7, u = i & 127;
    sXS[m][u] = x[(size_t)sS[m] * 128 + u];
  }
  // density = tanh((ef . Wd)^2) scattered into rho
  if (lane < 16) {
    float s = 0.f;
#pragma unroll
    for (int j = 0; j < 8; ++j) s += sEF[lane][j] * Wd[j];
    unsafeAtomicAdd(&rho[sR[lane]], tanhf(s * s));
  }
  __syncthreads();

  const int l16 = lane & 15;
  const int half = lane >> 4;

  // ---- layer 1: [16,8] @ Wr1[8,64], SiLU -> sHA ----
  for (int nt = 0; nt < 4; ++nt) {
    v8f c = v8f_zero();
#pragma unroll
    for (int k0 = 0; k0 < 8; k0 += 4) {
      const int u0 = k0 + 2 * half;
      v2f a = { sEF[l16][u0], sEF[l16][u0 + 1] };
      v2f b = { Wr1[u0 * 64 + nt * 16 + l16], Wr1[(u0 + 1) * 64 + nt * 16 + l16] };
      c = wmma_f32x4(a, b, c);
    }
#pragma unroll
    for (int i = 0; i < 8; ++i)
      sHA[i + 8 * half][nt * 16 + l16] = silu(c[i] * RS_8);
  }
  __syncthreads();

  // ---- layer 2: sHA @ Wr2[64,64], SiLU -> sHB ----
  for (int nt = 0; nt < 4; ++nt) {
    v8f c = v8f_zero();
    for (int k0 = 0; k0 < 64; k0 += 4) {
      const int u0 = k0 + 2 * half;
      v2f a = { sHA[l16][u0], sHA[l16][u0 + 1] };
      v2f b = { Wr2[u0 * 64 + nt * 16 + l16], Wr2[(u0 + 1) * 64 + nt * 16 + l16] };
      c = wmma_f32x4(a, b, c);
    }
#pragma unroll
    for (int i = 0; i < 8; ++i)
      sHB[i + 8 * half][nt * 16 + l16] = silu(c[i] * RS_64);
  }
  __syncthreads();

  // ---- layer 3: sHB @ Wr3[64,64], SiLU -> sHA ----
  for (int nt = 0; nt < 4; ++nt) {
    v8f c = v8f_zero();
    for (int k0 = 0; k0 < 64; k0 += 4) {
      const int u0 = k0 + 2 * half;
      v2f a = { sHB[l16][u0], sHB[l16][u0 + 1] };
      v2f b = { Wr3[u0 * 64 + nt * 16 + l16], Wr3[(u0 + 1) * 64 + nt * 16 + l16] };
      c = wmma_f32x4(a, b, c);
    }
#pragma unroll
    for (int i = 0; i < 8; ++i)
      sHA[i + 8 * half][nt * 16 + l16] = silu(c[i] * RS_64);
  }
  __syncthreads();

  // ---- layer 4: sHA @ Wr4[64,512] fused with tensor-product scatter ----
  for (int nt = 0; nt < 32; ++nt) {
    v8f c = v8f_zero();
    for (int k0 = 0; k0 < 64; k0 += 4) {
      const int u0 = k0 + 2 * half;
      v2f a = { sHA[l16][u0], sHA[l16][u0 + 1] };
      v2f b = { Wr4[u0 * 512 + nt * 16 + l16], Wr4[(u0 + 1) * 512 + nt * 16 + l16] };
      c = wmma_f32x4(a, b, c);
    }
    // fragment columns nt*16..nt*16+15 all belong to one l (128 % 16 == 0)
    const int cg = nt * 16 + l16;      // column in tp_w [4*128]
    const int l  = cg >> 7;            // irrep index
    const int u  = cg & 127;           // channel
    const int dcur = 2 * l + 1;
    const int off  = l * l;            // SH_OFF[l] = l^2
    const int base = 128 * off;
#pragma unroll
    for (int i = 0; i < 8; ++i) {
      const int m = i + 8 * half;      // edge within tile
      const float val = c[i] * RS_64 * sXS[m][u];
      float* row = msg + (size_t)sR[m] * MSG_W + base + u * dcur;
      for (int k = 0; k < dcur; ++k)
        unsafeAtomicAdd(row + k, val * sSH[m][off + k]);
    }
  }
}

// ---------------------------------------------------------------------------
// K3: message /= (rho+1); message[:, :128] += sinusoidal(rho) @ Wsin
// ---------------------------------------------------------------------------
__global__ __launch_bounds__(256) void norm_demb(
    float* __restrict__ msg, const float* __restrict__ rho,
    const float* __restrict__ Wsin) {
  const int n = blockIdx.x;
  const int t = threadIdx.x;
  __shared__ float se[32];

  const float r = rho[n];
  if (t < 32) {
    const int j = t & 15;
    const float f = __expf(-logf(100.0f) * (float)j / 16.0f);  // MAX_DENSITY=100
    const float ang = r * f;
    se[t] = (t < 16) ? __sinf(ang) : __cosf(ang);
  }
  __syncthreads();

  const float inv = 1.0f / (r + 1.0f);
  float* row = msg + (size_t)n * MSG_W;
  for (int c = t; c < MSG_W; c += 256) row[c] *= inv;

  if (t < 128) {
    float acc = 0.f;
#pragma unroll
    for (int j = 0; j < 32; ++j) acc += se[j] * Wsin[j * 128 + t];
    row[t] += acc;   // same thread that normalized row[t]; no race
  }
}

// ---------------------------------------------------------------------------
// K6: per-(node,l) skip contraction:
//   out[n,w,off+k] = (1/sqrt(1280)) * sum_v blk2[n,v,k] * sum_a attrs[n,a]*Wskip[l,v,a,w]
// Templated on L so per-thread accumulators (d registers) stay in VGPRs.
// ---------------------------------------------------------------------------
template <int L>
__global__ __launch_bounds__(128) void skip_out(
    const float* __restrict__ blk2, const float* __restrict__ attrs,
    const float* __restrict__ Wskip, float* __restrict__ out) {
  constexpr int D = 2 * L + 1;
  constexpr int OFF = L * L;
  constexpr int BASE = 128 * OFF;

  const int n = blockIdx.x;
  const int w = threadIdx.x;   // 128 threads = 128 output channels

  __shared__ float sA[N_SPECIES];
  __shared__ float sB[128 * D];

  if (w < N_SPECIES) sA[w] = attrs[(size_t)n * N_SPECIES + w];
  const float* __restrict__ src = blk2 + (size_t)n * MSG_W + BASE;
  for (int i = w; i < 128 * D; i += 128) sB[i] = src[i];
  __syncthreads();

  float acc[D];
#pragma unroll
  for (int k = 0; k < D; ++k) acc[k] = 0.f;

  const float* __restrict__ wb =
      Wskip + (size_t)L * 128 * N_SPECIES * 128 + w;
  for (int v = 0; v < 128; ++v) {
    const float* __restrict__ wv = wb + (size_t)v * (N_SPECIES * 128);
    float m = 0.f;
#pragma unroll
    for (int a = 0; a < N_SPECIES; ++a) m += sA[a] * wv[a * 128];
    const float* bv = &sB[v * D];
#pragma unroll
    for (int k = 0; k < D; ++k) acc[k] += bv[k] * m;
  }

  float* orow = out + (size_t)n * MSG_W + (size_t)w * 16 + OFF;
#pragma unroll
  for (int k = 0; k < D; ++k) orow[k] = acc[k] * RS_SKIP;
}

// ---------------------------------------------------------------------------
// Host-side launcher
// ---------------------------------------------------------------------------
extern "C" void kernel_launch(void* const* d_in, const int* in_sizes, int n_in,
                              void* d_out, int out_size, void* d_ws, size_t ws_size,
                              hipStream_t stream) {
  (void)in_sizes; (void)n_in; (void)out_size; (void)ws_size;

  const float* node_attrs = (const float*)d_in[0];
  const float* node_feats = (const float*)d_in[1];
  const float* edge_attrs = (const float*)d_in[2];
  const float* edge_feats = (const float*)d_in[3];
  const int*   edge_index = (const int*)d_in[4];
  const float* W_up  = (const float*)d_in[5];
  const float* Wr1   = (const float*)d_in[6];
  const float* Wr2   = (const float*)d_in[7];
  const float* Wr3   = (const float*)d_in[8];
  const float* Wr4   = (const float*)d_in[9];
  const float* Wd    = (const float*)d_in[10];
  const float* Wsin  = (const float*)d_in[11];
  const float* Wlin  = (const float*)d_in[12];
  const float* Wskip = (const float*)d_in[13];
  float* out = (float*)d_out;

  // Workspace layout (bytes), all 256B aligned:
  //   x    : [N,128]   =  5,120,000
  //   msg  : [N,2048]  = 81,920,000   (L2-resident accumulator)
  //   blk2 : [N,2048]  = 81,920,000
  //   rho  : [N]       =     40,000
  char* ws = (char*)d_ws;
  float* x    = (float*)(ws);
  float* msg  = (float*)(ws + 5120000);
  float* blk2 = (float*)(ws + 5120000 + 81920000);
  float* rhoB = (float*)(ws + 5120000 + 81920000 + 81920000);

  // K0: zero accumulators
  zero_f32<<<2048, 256, 0, stream>>>(msg, (size_t)N_NODES * MSG_W);
  zero_f32<<<64, 256, 0, stream>>>(rhoB, (size_t)N_NODES);

  // K1: x = node_feats @ (W_up / sqrt(128))   [10000,128]@[128,128]
  gemm_wmma_strided<1><<<dim3(N_NODES / 16, 8), 32, 0, stream>>>(
      node_feats, W_up, x, /*base=*/0, 128, 128, RS_MUL);

  // K2: fused radial MLP + density + tensor-product scatter (atomic into msg)
  radial_scatter<<<N_EDGES / 16, 32, 0, stream>>>(
      edge_feats, edge_attrs, edge_index, x, Wr1, Wr2, Wr3, Wr4, Wd, msg, rhoB);

  // K3: normalize by (rho+1) and add sinusoidal density embedding
  norm_demb<<<N_NODES, 256, 0, stream>>>(msg, rhoB, Wsin);

  // K5: per-l Wlin einsum as strided WMMA GEMMs: [N*d,128]@[128,128]
  gemm_wmma_strided<1><<<dim3((N_NODES * 1) / 16, 8), 32, 0, stream>>>(
      msg, Wlin + 0 * 128 * 128, blk2, /*base=*/128 * 0, MSG_W, MSG_W, RS_MUL);
  gemm_wmma_strided<3><<<dim3((N_NODES * 3) / 16, 8), 32, 0, stream>>>(
      msg, Wlin + 1 * 128 * 128, blk2, /*base=*/128 * 1, MSG_W, MSG_W, RS_MUL);
  gemm_wmma_strided<5><<<dim3((N_NODES * 5) / 16, 8), 32, 0, stream>>>(
      msg, Wlin + 2 * 128 * 128, blk2, /*base=*/128 * 4, MSG_W, MSG_W, RS_MUL);
  gemm_wmma_strided<7><<<dim3((N_NODES * 7) / 16, 8), 32, 0, stream>>>(
      msg, Wlin + 3 * 128 * 128, blk2, /*base=*/128 * 9, MSG_W, MSG_W, RS_MUL);

  // K6: species-mixed skip contraction -> output [N,128,16]
  skip_out<0><<<N_NODES, 128, 0, stream>>>(blk2, node_attrs, Wskip, out);
  skip_out<1><<<N_NODES, 128, 0, stream>>>(blk2, node_attrs, Wskip, out);
  skip_out<2><<<N_NODES, 128, 0, stream>>>(blk2, node_attrs, Wskip, out);
  skip_out<3><<<N_NODES, 128, 0, stream>>>(blk2, node_attrs, Wskip, out);
}